// SyntheticVisionTower_20418274525913
// MI455X (gfx1250) — compile-verified
//
#include <hip/hip_runtime.h>

typedef __attribute__((ext_vector_type(2))) float v2f;
typedef __attribute__((ext_vector_type(8))) float v8f;

#define HIDDEN 4096
#define MMK    7
#define MROWS  36864   // 64 * 576
#define MT_PER_WAVE 4

// Out[M, 4096] = X[M, 7] @ A[0:7, 0:4096]   (fp32, via V_WMMA_F32_16X16X4_F32)
__global__ __launch_bounds__(256) void vision_proj_wmma(
    const float* __restrict__ X,   // [MROWS, 7]
    const float* __restrict__ A,   // [4096, 4096], we use rows 0..6
    float* __restrict__ Out)       // [MROWS, 4096]
{
    const int lane = threadIdx.x & 31;
    const int wave = threadIdx.x >> 5;
    const int lan  = lane & 15;   // position within half-wave
    const int half = lane >> 4;   // 0: K pair {0,1}/{4,5}; 1: K pair {2,3}/{6,-}

    const int n0 = blockIdx.x * 16;           // N tile (one per block, shared by 8 waves)

    // ---- B fragments (A[:7, n0:n0+16]) -- loaded once, reused across 4 M tiles ----
    // wmma #1 covers K=0..3, wmma #2 covers K=4..6 (+ zero pad at K=7).
    const int   kA  = half * 2;               // 0 or 2
    const float sel = half ? 0.0f : 1.0f;     // kills the padded K=7 element
    const float* wc = A + n0 + lan;           // column n0+lan, row stride HIDDEN
    v2f b0, b1;
    b0.x = wc[(size_t)(kA    ) * HIDDEN];
    b0.y = wc[(size_t)(kA + 1) * HIDDEN];
    b1.x = wc[(size_t)(kA + 4) * HIDDEN];             // K=4 or K=6
    b1.y = wc[(size_t)(5 + half) * HIDDEN] * sel;     // K=5, or K=6*0 (safe, in-bounds)

    const int mt_base = (blockIdx.y * 8 + wave) * MT_PER_WAVE;

#pragma unroll
    for (int i = 0; i < MT_PER_WAVE; ++i) {
        const int m0  = (mt_base + i) * 16;
        const int row = m0 + lan;
        const float* xr = X + (size_t)row * MMK;

        // ---- A fragments (X[m0:m0+16, 0:7]) ----
        v2f a0, a1;
        a0.x = xr[kA];
        a0.y = xr[kA + 1];
        a1.x = xr[kA + 4];                    // K=4 or K=6
        a1.y = xr[5 + half] * sel;            // K=5, or K=6*0 (never reads index 7)

        v8f c = {};
        c = __builtin_amdgcn_wmma_f32_16x16x4_f32(
                false, a0, false, b0, (short)0, c, false, false);
        c = __builtin_amdgcn_wmma_f32_16x16x4_f32(
                false, a1, false, b1, (short)0, c, false, false);

        // ---- C store: VGPR r -> row m0+r (lanes 0-15) / m0+8+r (lanes 16-31) ----
        // Non-temporal: 604 MB streamed out once; keep L2 for X and A.
        float* orow = Out + (size_t)(m0 + half * 8) * HIDDEN + n0 + lan;
#pragma unroll
        for (int r = 0; r < 8; ++r) {
            __builtin_nontemporal_store(c[r], orow + (size_t)r * HIDDEN);
        }
    }
}

extern "C" void kernel_launch(void* const* d_in, const int* in_sizes, int n_in,
                              void* d_out, int out_size, void* d_ws, size_t ws_size,
                              hipStream_t stream) {
    (void)in_sizes; (void)n_in; (void)out_size; (void)d_ws; (void)ws_size;
    const float* images = (const float*)d_in[0];   // [64, 576, 7]
    const float* A      = (const float*)d_in[1];   // [4096, 4096]
    float*       out    = (float*)d_out;           // [64, 576, 4096]

    dim3 grid(HIDDEN / 16,                      // 256 N tiles
              MROWS / (16 * 8 * MT_PER_WAVE)); // 72 M groups (8 waves x 4 tiles x 16 rows)
    vision_proj_wmma<<<grid, 256, 0, stream>>>(images, A, out);
}